// GaussianHMM_75316546502843
// MI455X (gfx1250) — compile-verified
//
#include <hip/hip_runtime.h>
#include <hip/hip_bf16.h>
#include <math.h>

// ---------------------------------------------------------------------------
// GaussianHMM on MI455X (gfx1250, wave32):
//   K1 setup   : per-state Cholesky, W = L^-1, u = W*mu, log-det consts,
//                expA = softmax rows of log_A, pinit = softmax(log_pi)
//   K2 emit    : log_em[T,K] via f32 WMMA GEMM  Z = Obs * Wstack^T  (obs tile
//                staged to LDS with global_load_async_to_lds_b128; fused
//                u-subtract / square / 16-lane reduce)
//   K3 chunkmat: per-64-step transfer matrices G_c = prod(D_t * expA) via
//                f32 WMMA 32x32 matmuls, max-rescaled each step
//   K4 bound   : sequential scan over 1024 chunk matrices (tiny)
//   K5 scan    : within-chunk replay, writes normalized p_t == softmax output
// ---------------------------------------------------------------------------

#define T_TOT   65536
#define KST     32
#define DDIM    32
#define NCHUNK  1024
#define CLEN    64
#define LOG_TAU 1.8378770664093453f

typedef __attribute__((ext_vector_type(2))) float v2f;
typedef __attribute__((ext_vector_type(8))) float v8f;
typedef __attribute__((ext_vector_type(4))) int   v4i;

typedef __attribute__((address_space(1))) v4i* glb_b128_t;  // global int4*
typedef __attribute__((address_space(3))) v4i* lds_b128_t;  // LDS int4*

__device__ __forceinline__ v8f wmma_f32_k4(v2f a, v2f b, v8f c) {
  // D = A(16x4,f32) * B(4x16,f32) + C(16x16,f32)   [VOP3P, full wave32]
  return __builtin_amdgcn_wmma_f32_16x16x4_f32(
      /*neg_a=*/false, a, /*neg_b=*/false, b,
      /*c_mod=*/(short)0, c, /*reuse_a=*/false, /*reuse_b=*/false);
}

__device__ __forceinline__ float wsum32(float v) {
  #pragma unroll
  for (int m = 16; m >= 1; m >>= 1) v += __shfl_xor(v, m, 32);
  return v;
}
__device__ __forceinline__ float wmax32(float v) {
  #pragma unroll
  for (int m = 16; m >= 1; m >>= 1) v = fmaxf(v, __shfl_xor(v, m, 32));
  return v;
}
// sum within each 16-lane half of the wave
__device__ __forceinline__ float hsum16(float v) {
  #pragma unroll
  for (int m = 8; m >= 1; m >>= 1) v += __shfl_xor(v, m, 32);
  return v;
}

// ---------------------------------------------------------------------------
// K1: one block (32 threads = 1 wave) per state k
// ---------------------------------------------------------------------------
__global__ void hmm_setup_kernel(const float* __restrict__ chol,
                                 const float* __restrict__ mean,
                                 const float* __restrict__ log_pi,
                                 const float* __restrict__ log_A,
                                 float* __restrict__ BT,     // [32][1024]  B^T for GEMM
                                 float* __restrict__ u,      // [K*32]
                                 float* __restrict__ ck,     // [K] emission consts
                                 float* __restrict__ expA,   // [K][K] transition probs
                                 float* __restrict__ pinit)  // [K]
{
  __shared__ float L[DDIM][DDIM];
  const int lane = threadIdx.x;
  const int k = blockIdx.x;
  const float* Ck = chol + k * DDIM * DDIM;

  // cov = chol * chol^T + 1e-6 I  (lane = row)
  float ri[DDIM];
  #pragma unroll
  for (int m = 0; m < DDIM; ++m) ri[m] = Ck[lane * DDIM + m];
  for (int j = 0; j < DDIM; ++j) {
    float s = (lane == j) ? 1e-6f : 0.0f;
    #pragma unroll
    for (int m = 0; m < DDIM; ++m) s = fmaf(ri[m], Ck[j * DDIM + m], s);
    L[lane][j] = s;
  }
  __syncthreads();

  // in-place Cholesky (left-looking over columns)
  for (int j = 0; j < DDIM; ++j) {
    float acc = L[lane][j];
    for (int m = 0; m < j; ++m) acc -= L[lane][m] * L[j][m];
    if (lane == j) L[j][j] = sqrtf(acc);
    __syncthreads();
    if (lane > j) L[lane][j] = acc / L[j][j];
    __syncthreads();
  }

  // emission constant: -0.5*(D*log(2pi) + 2*sum(log diag))
  float sld = wsum32(logf(L[lane][lane]));
  if (lane == 0) ck[k] = -0.5f * ((float)DDIM * LOG_TAU + 2.0f * sld);

  // W = L^-1, column c = lane via forward substitution (registers)
  float x[DDIM];
  #pragma unroll
  for (int i = 0; i < DDIM; ++i) {
    float s = (i == lane) ? 1.0f : 0.0f;
    #pragma unroll
    for (int m = 0; m < i; ++m) s -= L[i][m] * x[m];
    x[i] = s / L[i][i];     // exact 0 above the diagonal
  }

  // BT[j][k*32+i] = W[i][j]   (j = lane)
  #pragma unroll
  for (int i = 0; i < DDIM; ++i) BT[lane * (KST * DDIM) + k * DDIM + i] = x[i];

  // u[k*32+i] = sum_j W[i][j] * mean[k][j]
  float mval = mean[k * DDIM + lane];
  #pragma unroll
  for (int i = 0; i < DDIM; ++i) {
    float ui = wsum32(x[i] * mval);
    if (lane == i) u[k * DDIM + i] = ui;
  }

  // expA row k = softmax(log_A_raw[k,:])
  float va = log_A[k * KST + lane];
  float ea = expf(va - wmax32(va));
  expA[k * KST + lane] = ea / wsum32(ea);

  if (k == 0) {
    float vp = log_pi[lane];
    float ep = expf(vp - wmax32(vp));
    pinit[lane] = ep / wsum32(ep);
  }
}

// ---------------------------------------------------------------------------
// K2: emissions. 1024 blocks x 256 threads (8 waves). Block = 64 time rows.
// obs tile (8KB, contiguous) staged to LDS with async-to-LDS loads.
// Wave -> 16 rows x 512 columns (16 states). WMMA f32 16x16x4, K-chain of 8.
// ---------------------------------------------------------------------------
__global__ void hmm_emit_kernel(const float* __restrict__ obs,
                                const float* __restrict__ BT,
                                const float* __restrict__ u,
                                const float* __restrict__ ck,
                                float* __restrict__ log_em)
{
  __shared__ float obs_s[64 * DDIM];   // 8 KB: 64 time rows x 32 features
  const int tid  = threadIdx.x;
  const int lane = threadIdx.x & 31;
  const int wv   = threadIdx.x >> 5;   // 0..7
  const int rg   = wv & 3;             // row group (16 rows)
  const int chf  = wv >> 2;            // column half (16 states)
  const int t0   = blockIdx.x * 64 + rg * 16;
  const int mrow = lane & 15;
  const int koff = (lane < 16) ? 0 : 2;

  // ---- stage the 64x32 obs block (linear 8KB) into LDS ----
  {
    const float* gsrc = obs + (size_t)blockIdx.x * 64 * DDIM;
#if __has_builtin(__builtin_amdgcn_global_load_async_to_lds_b128)
    // 2 x 16B per thread, tracked by ASYNCcnt
    __builtin_amdgcn_global_load_async_to_lds_b128(
        (glb_b128_t)(gsrc + tid * 4),
        (lds_b128_t)(obs_s + tid * 4), 0, 0);
    __builtin_amdgcn_global_load_async_to_lds_b128(
        (glb_b128_t)(gsrc + 1024 + tid * 4),
        (lds_b128_t)(obs_s + 1024 + tid * 4), 0, 0);
#if __has_builtin(__builtin_amdgcn_s_wait_asynccnt)
    __builtin_amdgcn_s_wait_asynccnt(0);
#else
    asm volatile("s_wait_asynccnt 0" ::: "memory");
#endif
#else
    // fallback: synchronous vectorized stage
    float4 a0 = ((const float4*)gsrc)[tid];
    float4 a1 = ((const float4*)gsrc)[tid + 256];
    ((float4*)obs_s)[tid] = a0;
    ((float4*)obs_s)[tid + 256] = a1;
#endif
  }
  __syncthreads();   // every wave has drained its own async copies first

  if (blockIdx.x + 1 < gridDim.x)
    __builtin_prefetch(obs + ((size_t)(blockIdx.x + 1) * 64 + tid) * DDIM, 0, 0);

  // A operand from LDS: obs rows, K-chunks of 4 (VGPR0 = K0/K2, VGPR1 = K1/K3)
  const int rloc = rg * 16 + mrow;
  v2f a[8];
  #pragma unroll
  for (int c = 0; c < 8; ++c) {
    v2f t;
    t.x = obs_s[rloc * DDIM + 4 * c + koff];
    t.y = obs_s[rloc * DDIM + 4 * c + koff + 1];
    a[c] = t;
  }

  #pragma unroll 1
  for (int s = 0; s < 16; ++s) {
    const int k = chf * 16 + s;
    float mh[8];
    #pragma unroll
    for (int r = 0; r < 8; ++r) mh[r] = 0.0f;

    #pragma unroll
    for (int h = 0; h < 2; ++h) {
      const int n0 = k * DDIM + h * 16;
      v8f acc = {0.f, 0.f, 0.f, 0.f, 0.f, 0.f, 0.f, 0.f};
      #pragma unroll
      for (int c = 0; c < 8; ++c) {
        const float* bp = BT + (4 * c + koff) * (KST * DDIM) + n0 + mrow;
        v2f b; b.x = bp[0]; b.y = bp[KST * DDIM];
        acc = wmma_f32_k4(a[c], b, acc);
      }
      const float uu = u[n0 + mrow];
      #pragma unroll
      for (int r = 0; r < 8; ++r) {
        float d = acc[r] - uu;
        mh[r] += hsum16(d * d);   // reduce over 16 columns of this half
      }
    }
    const float c0 = ck[k];
    if (mrow == 0) {              // lanes 0 (rows r) and 16 (rows r+8)
      const int rbase = t0 + ((lane >> 4) << 3);
      #pragma unroll
      for (int r = 0; r < 8; ++r)
        log_em[(rbase + r) * KST + k] = c0 - 0.5f * mh[r];
    }
  }
}

// ---------------------------------------------------------------------------
// K3: chunk transfer matrices. 1024 blocks x 1 wave.
// M <- diag(w_t) * expA * M, rescaled by max each step; 32x32 via 4 WMMA tiles.
// Stores G transposed for coalesced access in K4.
// ---------------------------------------------------------------------------
__global__ void hmm_chunkmat_kernel(const float* __restrict__ log_em,
                                    const float* __restrict__ expA,
                                    float* __restrict__ Gt)
{
  __shared__ float M[2][KST * KST];
  const int lane  = threadIdx.x;
  const int blk   = blockIdx.x;
  const int mrow  = lane & 15;
  const int koff  = (lane < 16) ? 0 : 2;
  const int half8 = (lane >> 4) << 3;

  // Preload A operand tiles of expA: row tiles R in {0,1}, K-chunks 0..7
  v2f aE[2][8];
  #pragma unroll
  for (int R = 0; R < 2; ++R)
    #pragma unroll
    for (int c = 0; c < 8; ++c) {
      const float* p = expA + (R * 16 + mrow) * KST + 4 * c + koff;
      v2f t; t.x = p[0]; t.y = p[1];
      aE[R][c] = t;
    }

  int tstart = blk * CLEN;
  if (blk == 0) {
    // M = diag(w_0) : absorbs the init step p_0 = pinit .* w_0 (K4 starts from pinit)
    float le = log_em[lane];
    float w = expf(le - wmax32(le));
    for (int j = 0; j < KST; ++j) M[0][lane * KST + j] = (j == lane) ? w : 0.0f;
    tstart = 1;
  } else {
    for (int j = 0; j < KST; ++j) M[0][lane * KST + j] = (j == lane) ? 1.0f : 0.0f;
  }
  __syncthreads();

  int cur = 0;
  const int tend = blk * CLEN + CLEN;
  for (int t = tstart; t < tend; ++t) {
    float le = log_em[t * KST + lane];
    float w = expf(le - wmax32(le));

    float dreg[2][2][8];
    #pragma unroll
    for (int R = 0; R < 2; ++R)
      #pragma unroll
      for (int Cc = 0; Cc < 2; ++Cc) {
        v8f acc = {0.f, 0.f, 0.f, 0.f, 0.f, 0.f, 0.f, 0.f};
        #pragma unroll
        for (int c = 0; c < 8; ++c) {
          v2f b;
          b.x = M[cur][(4 * c + koff) * KST + Cc * 16 + mrow];
          b.y = M[cur][(4 * c + koff + 1) * KST + Cc * 16 + mrow];
          acc = wmma_f32_k4(aE[R][c], b, acc);
        }
        #pragma unroll
        for (int r = 0; r < 8; ++r) {
          int rowi = R * 16 + r + half8;
          dreg[R][Cc][r] = acc[r] * __shfl(w, rowi, 32);  // row scale by w_t
        }
      }

    // global max rescale (entries are nonnegative, max > 0)
    float mx = 0.0f;
    #pragma unroll
    for (int R = 0; R < 2; ++R)
      #pragma unroll
      for (int Cc = 0; Cc < 2; ++Cc)
        #pragma unroll
        for (int r = 0; r < 8; ++r) mx = fmaxf(mx, dreg[R][Cc][r]);
    mx = wmax32(mx);
    const float inv = 1.0f / mx;

    const int nxt = cur ^ 1;
    #pragma unroll
    for (int R = 0; R < 2; ++R)
      #pragma unroll
      for (int Cc = 0; Cc < 2; ++Cc)
        #pragma unroll
        for (int r = 0; r < 8; ++r)
          M[nxt][(R * 16 + r + half8) * KST + Cc * 16 + mrow] = dreg[R][Cc][r] * inv;
    cur = nxt;
    __syncthreads();
  }

  // Gt[blk][j][i] = M[i][j]  (transposed for coalesced matvec in K4)
  for (int i = 0; i < KST; ++i)
    Gt[blk * (KST * KST) + lane * KST + i] = M[cur][i * KST + lane];
}

// ---------------------------------------------------------------------------
// K4: sequential boundary scan over 1024 chunk matrices (1 wave)
// ---------------------------------------------------------------------------
__global__ void hmm_bound_kernel(const float* __restrict__ Gt,
                                 const float* __restrict__ pinit,
                                 float* __restrict__ Pbound)
{
  const int lane = threadIdx.x;
  float p = pinit[lane];
  for (int c = 0; c < NCHUNK; ++c) {
    Pbound[c * KST + lane] = p;            // incoming state of chunk c
    const float* g = Gt + c * (KST * KST);
    float acc = 0.0f;
    #pragma unroll
    for (int j = 0; j < KST; ++j)
      acc = fmaf(g[j * KST + lane], __shfl(p, j, 32), acc);  // (G p)[lane]
    p = acc / wsum32(acc);
  }
}

// ---------------------------------------------------------------------------
// K5: within-chunk replay; p_t is the soft path (softmax of log_deltas)
// ---------------------------------------------------------------------------
__global__ void hmm_scan_kernel(const float* __restrict__ log_em,
                                const float* __restrict__ expA,
                                const float* __restrict__ Pbound,
                                float* __restrict__ out)
{
  const int lane = threadIdx.x;
  const int c = blockIdx.x;
  float arow[KST];
  #pragma unroll
  for (int j = 0; j < KST; ++j) arow[j] = expA[lane * KST + j];

  float p = Pbound[c * KST + lane];
  int t = c * CLEN;
  const int tend = t + CLEN;
  if (c == 0) {
    float le = log_em[lane];
    float w = expf(le - wmax32(le));
    float v = p * w;                       // pinit .* w_0
    p = v / wsum32(v);
    out[lane] = p;
    t = 1;
  }
  for (; t < tend; ++t) {
    float acc = 0.0f;
    #pragma unroll
    for (int j = 0; j < KST; ++j)
      acc = fmaf(arow[j], __shfl(p, j, 32), acc);   // expA * p
    float le = log_em[t * KST + lane];
    float w = expf(le - wmax32(le));
    float v = acc * w;
    p = v / wsum32(v);
    out[t * KST + lane] = p;
  }
}

// ---------------------------------------------------------------------------
extern "C" void kernel_launch(void* const* d_in, const int* in_sizes, int n_in,
                              void* d_out, int out_size, void* d_ws, size_t ws_size,
                              hipStream_t stream) {
  const float* obs    = (const float*)d_in[0];  // [T,D]
  const float* mean   = (const float*)d_in[1];  // [K,D]
  const float* chol   = (const float*)d_in[2];  // [K,D,D]
  const float* log_pi = (const float*)d_in[3];  // [K]
  const float* log_A  = (const float*)d_in[4];  // [K,K]
  float* out = (float*)d_out;                   // [T,K]

  float* ws      = (float*)d_ws;
  float* log_em  = ws;                                   // T*K      = 2,097,152
  float* BT      = log_em + (size_t)T_TOT * KST;         // 32*1024  = 32,768
  float* u       = BT + (size_t)DDIM * KST * DDIM;       // 1024
  float* ck      = u + KST * DDIM;                       // 32
  float* expA    = ck + KST;                             // 1024
  float* pinit   = expA + KST * KST;                     // 32
  float* Gt      = pinit + KST;                          // 1024*1024 = 1,048,576
  float* Pbound  = Gt + (size_t)NCHUNK * KST * KST;      // 1024*32   = 32,768

  hmm_setup_kernel<<<KST, 32, 0, stream>>>(chol, mean, log_pi, log_A,
                                           BT, u, ck, expA, pinit);
  hmm_emit_kernel<<<T_TOT / 64, 256, 0, stream>>>(obs, BT, u, ck, log_em);
  hmm_chunkmat_kernel<<<NCHUNK, 32, 0, stream>>>(log_em, expA, Gt);
  hmm_bound_kernel<<<1, 32, 0, stream>>>(Gt, pinit, Pbound);
  hmm_scan_kernel<<<NCHUNK, 32, 0, stream>>>(log_em, expA, Pbound, out);
}